// Generator_86895778332853
// MI455X (gfx1250) — compile-verified
//
#include <hip/hip_runtime.h>
#include <math.h>

typedef __attribute__((ext_vector_type(2))) float v2f;
typedef __attribute__((ext_vector_type(8))) float v8f;

#define BB 64
#define TT 32
#define EE 100
#define DYY 200
#define DZZ 500
#define HH 700
#define VV 32000
#define TM1 31
#define GAMMA_F 0.5f
#define EPS_F 1e-10f
#define SOS_ID 2

// ---------------------------------------------------------------------------
// init: h0 = [ (1-labels) W_fc^T + b_fc , z ],  x0 = embed[SOS], zero pads
// ---------------------------------------------------------------------------
__global__ void gen_init_kernel(const float* __restrict__ z,
                                const float* __restrict__ labels,
                                const float* __restrict__ W_fc,
                                const float* __restrict__ b_fc,
                                const float* __restrict__ embed,
                                float* __restrict__ out,
                                float* __restrict__ hbuf0,
                                float* __restrict__ xbuf0) {
  const long n_pred0 = (long)BB * VV;   // predictions[0] = 0
  const long n_out1  = (long)BB * HH;   // outputs[1] = 0
  const long n_h0    = (long)BB * HH;   // h0 -> outputs[0] and hbuf0
  const long n_x0    = (long)BB * EE;   // x0 -> xbuf0
  const long total   = n_pred0 + n_out1 + n_h0 + n_x0;
  float* predictions = out + (long)(TT + 1) * BB * HH;
  long i = (long)blockIdx.x * blockDim.x + threadIdx.x;
  const long stride = (long)gridDim.x * blockDim.x;
  for (; i < total; i += stride) {
    if (i < n_pred0) {
      predictions[i] = 0.f;
    } else if (i < n_pred0 + n_out1) {
      out[(long)BB * HH + (i - n_pred0)] = 0.f;
    } else if (i < n_pred0 + n_out1 + n_h0) {
      long j = i - n_pred0 - n_out1;
      int b = (int)(j / HH), jj = (int)(j % HH);
      float v = (jj < DYY) ? ((1.f - labels[b]) * W_fc[jj] + b_fc[jj])
                           : z[(long)b * DZZ + (jj - DYY)];
      hbuf0[j] = v;
      out[j] = v;                       // outputs[0]
    } else {
      long j = i - n_pred0 - n_out1 - n_h0;
      int e = (int)(j % EE);
      xbuf0[j] = embed[(long)SOS_ID * EE + e];
    }
  }
}

// ---------------------------------------------------------------------------
// GRU cell: one block per batch row, x/h staged in LDS
// ---------------------------------------------------------------------------
__global__ void __launch_bounds__(256)
gen_gru_kernel(const float* __restrict__ h_in, const float* __restrict__ x_in,
               const float* __restrict__ W_ih, const float* __restrict__ W_hh,
               const float* __restrict__ b_ih, const float* __restrict__ b_hh,
               float* __restrict__ h_out, float* __restrict__ out_slot) {
  __shared__ float xs[EE];
  __shared__ float hs[HH];
  const int b = blockIdx.x;
  for (int i = threadIdx.x; i < EE; i += blockDim.x) xs[i] = x_in[(long)b * EE + i];
  for (int i = threadIdx.x; i < HH; i += blockDim.x) hs[i] = h_in[(long)b * HH + i];
  __syncthreads();
  for (int j = threadIdx.x; j < HH; j += blockDim.x) {
    float gi[3], gh[3];
#pragma unroll
    for (int g = 0; g < 3; ++g) {
      const float* wi = W_ih + (long)(g * HH + j) * EE;
      float a0 = b_ih[g * HH + j];
      for (int k = 0; k < EE; ++k) a0 = fmaf(xs[k], wi[k], a0);
      gi[g] = a0;
      const float* wh = W_hh + (long)(g * HH + j) * HH;
      float a1 = b_hh[g * HH + j];
      for (int k = 0; k < HH; ++k) a1 = fmaf(hs[k], wh[k], a1);
      gh[g] = a1;
    }
    float r  = 1.f / (1.f + __expf(-(gi[0] + gh[0])));
    float zt = 1.f / (1.f + __expf(-(gi[1] + gh[1])));
    float nn = tanhf(gi[2] + r * gh[2]);
    float hn = (1.f - zt) * nn + zt * hs[j];
    h_out[(long)b * HH + j] = hn;
    out_slot[(long)b * HH + j] = hn;     // outputs[t+2]
  }
}

// ---------------------------------------------------------------------------
// pred = h @ W_out^T + b_out   (WMMA f32 16x16x4)
// 8 waves/block, wave owns one 16-col N tile and all four 16-row M tiles.
// Named accumulators to avoid spurious acc copies; prefetch streams W_out.
// Fused: write pred to d_out, pred+gumbel to pg workspace.
// ---------------------------------------------------------------------------
__global__ void __launch_bounds__(256)
gen_pred_kernel(const float* __restrict__ h, const float* __restrict__ W_out,
                const float* __restrict__ b_out, const float* __restrict__ gu,
                float* __restrict__ pred_out, float* __restrict__ pg) {
  const int lane  = threadIdx.x & 31;
  const int wave  = threadIdx.x >> 5;
  const int nbase = (blockIdx.x * 8 + wave) * 16;
  const int idx15 = lane & 15;          // A: M row, B/D: N col
  const int khalf = lane >> 4;          // selects K pair {0,1} vs {2,3}

  v8f acc0 = {}, acc1 = {}, acc2 = {}, acc3 = {};
  // per-lane base pointers (hoisted)
  const float* brow = W_out + (long)(nbase + idx15) * HH + 2 * khalf; // B[k][n]=W_out[n][k]
  const float* arow = h + (long)idx15 * HH + 2 * khalf;

#pragma unroll 2
  for (int k0 = 0; k0 < HH; k0 += 4) {
    __builtin_prefetch(brow + k0 + 96, 0, 1);     // global_prefetch_b8 on W_out stream
    v2f bfrag = *(const v2f*)(brow + k0);
    v2f a0 = *(const v2f*)(arow + k0);
    v2f a1 = *(const v2f*)(arow + (long)16 * HH + k0);
    v2f a2 = *(const v2f*)(arow + (long)32 * HH + k0);
    v2f a3 = *(const v2f*)(arow + (long)48 * HH + k0);
    acc0 = __builtin_amdgcn_wmma_f32_16x16x4_f32(false, a0, false, bfrag, (short)0, acc0, false, false);
    acc1 = __builtin_amdgcn_wmma_f32_16x16x4_f32(false, a1, false, bfrag, (short)0, acc1, false, false);
    acc2 = __builtin_amdgcn_wmma_f32_16x16x4_f32(false, a2, false, bfrag, (short)0, acc2, false, false);
    acc3 = __builtin_amdgcn_wmma_f32_16x16x4_f32(false, a3, false, bfrag, (short)0, acc3, false, false);
  }

  const int n = nbase + idx15;
  const float bias = b_out[n];
  v8f accs[4] = {acc0, acc1, acc2, acc3};
#pragma unroll
  for (int mt = 0; mt < 4; ++mt) {
#pragma unroll
    for (int r = 0; r < 8; ++r) {
      const int row = mt * 16 + khalf * 8 + r;        // D layout, ISA 7.12.2
      const float p = accs[mt][r] + bias;
      pred_out[(long)row * VV + n] = p;
      const float u = gu[(long)row * VV + n];
      const float g = -__logf(-__logf(u + EPS_F) + EPS_F);
      pg[(long)row * VV + n] = p + g;
    }
  }
}

// ---------------------------------------------------------------------------
// softmax stats per batch row (max, 1/(sum*GAMMA)); also zero x accumulator
// ---------------------------------------------------------------------------
__global__ void __launch_bounds__(256)
gen_stats_kernel(const float* __restrict__ pg, float* __restrict__ stats,
                 float* __restrict__ xnext) {
  __shared__ float red[256];
  const int b = blockIdx.x;
  const float* row = pg + (long)b * VV;
  float m = -3.0e38f;
  for (int v = threadIdx.x; v < VV; v += 256) m = fmaxf(m, row[v]);
  red[threadIdx.x] = m; __syncthreads();
  for (int s = 128; s > 0; s >>= 1) {
    if (threadIdx.x < s) red[threadIdx.x] = fmaxf(red[threadIdx.x], red[threadIdx.x + s]);
    __syncthreads();
  }
  m = red[0]; __syncthreads();
  float sum = 0.f;
  for (int v = threadIdx.x; v < VV; v += 256) sum += __expf(row[v] - m);
  red[threadIdx.x] = sum; __syncthreads();
  for (int s = 128; s > 0; s >>= 1) {
    if (threadIdx.x < s) red[threadIdx.x] += red[threadIdx.x + s];
    __syncthreads();
  }
  if (threadIdx.x == 0) { stats[2 * b] = m; stats[2 * b + 1] = 1.f / (red[0] * GAMMA_F); }
  for (int e = threadIdx.x; e < EE; e += 256) xnext[(long)b * EE + e] = 0.f;
}

// ---------------------------------------------------------------------------
// x_next = (softmax(pg)/GAMMA) @ embed   (WMMA f32 16x16x4, K = V split x8)
// exp applied on-the-fly in the A fragment; row-uniform 1/(sum*GAMMA) scale
// applied once per output after accumulation; float atomics combine K splits.
// ---------------------------------------------------------------------------
#define KSPLIT 8
__global__ void __launch_bounds__(128)
gen_embed_kernel(const float* __restrict__ pg, const float* __restrict__ stats,
                 const float* __restrict__ embed, float* __restrict__ xnext) {
  const int lane  = threadIdx.x & 31;
  const int mt    = threadIdx.x >> 5;   // wave -> M tile
  const int nbase = blockIdx.x * 16;
  const int idx15 = lane & 15;
  const int khalf = lane >> 4;
  const int m = mt * 16 + idx15;        // A-fragment row (batch index)
  const int e = nbase + idx15;          // B-fragment col (embedding dim)
  const float rowmax = stats[2 * m];
  const int kchunk = VV / KSPLIT;
  const int kstart = blockIdx.y * kchunk;
  const int kend   = kstart + kchunk;
  const float* prow = pg + (long)m * VV + 2 * khalf;
  const float* ecol0 = embed + e + (long)2 * khalf * EE;
  const bool ein = (e < EE);

  v8f acc = {};
  for (int k0 = kstart; k0 < kend; k0 += 4) {
    __builtin_prefetch(prow + k0 + 64, 0, 1);
    v2f pv = *(const v2f*)(prow + k0);
    v2f afrag;
    afrag.x = __expf(pv.x - rowmax);
    afrag.y = __expf(pv.y - rowmax);
    v2f bfrag;
    bfrag.x = ein ? ecol0[(long)k0 * EE] : 0.f;
    bfrag.y = ein ? ecol0[(long)k0 * EE + EE] : 0.f;
    acc = __builtin_amdgcn_wmma_f32_16x16x4_f32(
        false, afrag, false, bfrag, (short)0, acc, false, false);
  }
  if (ein) {
#pragma unroll
    for (int r = 0; r < 8; ++r) {
      const int row = mt * 16 + khalf * 8 + r;        // D layout row
      atomicAdd(&xnext[(long)row * EE + e], acc[r] * stats[2 * row + 1]);
    }
  }
}

// ---------------------------------------------------------------------------
extern "C" void kernel_launch(void* const* d_in, const int* in_sizes, int n_in,
                              void* d_out, int out_size, void* d_ws, size_t ws_size,
                              hipStream_t stream) {
  (void)in_sizes; (void)n_in; (void)out_size; (void)ws_size;
  const float* z      = (const float*)d_in[0];
  const float* labels = (const float*)d_in[1];
  const float* W_fc   = (const float*)d_in[2];
  const float* b_fc   = (const float*)d_in[3];
  const float* embed  = (const float*)d_in[4];
  const float* W_ih   = (const float*)d_in[5];
  const float* W_hh   = (const float*)d_in[6];
  const float* b_ih   = (const float*)d_in[7];
  const float* b_hh   = (const float*)d_in[8];
  const float* W_out  = (const float*)d_in[9];
  const float* b_out  = (const float*)d_in[10];
  const float* gu     = (const float*)d_in[11];

  float* out = (float*)d_out;
  float* outputs     = out;                               // [T+1, B, H]
  float* predictions = out + (long)(TT + 1) * BB * HH;    // [T, B, V]

  float* ws    = (float*)d_ws;
  float* hbuf0 = ws;                         // B*H
  float* hbuf1 = hbuf0 + (long)BB * HH;      // B*H
  float* xbuf0 = hbuf1 + (long)BB * HH;      // B*E
  float* xbuf1 = xbuf0 + (long)BB * EE;      // B*E
  float* pg    = xbuf1 + (long)BB * EE;      // B*V  (pred + gumbel, L2-resident)
  float* stats = pg + (long)BB * VV;         // 2*B

  gen_init_kernel<<<512, 256, 0, stream>>>(z, labels, W_fc, b_fc, embed,
                                           out, hbuf0, xbuf0);

  float* hb[2] = {hbuf0, hbuf1};
  float* xb[2] = {xbuf0, xbuf1};
  for (int t = 0; t < TM1; ++t) {
    const float* h_in  = hb[t & 1];
    float*       h_out = hb[(t + 1) & 1];
    const float* x_in  = xb[t & 1];
    float*       x_out = xb[(t + 1) & 1];

    gen_gru_kernel<<<BB, 256, 0, stream>>>(h_in, x_in, W_ih, W_hh, b_ih, b_hh,
                                           h_out,
                                           outputs + (long)(t + 2) * BB * HH);
    gen_pred_kernel<<<VV / 128, 256, 0, stream>>>(h_out, W_out, b_out,
                                                  gu + (long)t * BB * VV,
                                                  predictions + (long)(t + 1) * BB * VV,
                                                  pg);
    gen_stats_kernel<<<BB, 256, 0, stream>>>(pg, stats, x_out);
    dim3 eg((EE + 15) / 16, KSPLIT);
    gen_embed_kernel<<<eg, 128, 0, stream>>>(pg, stats, embed, x_out);
  }
}